// LoRALayer_12884901888328
// MI455X (gfx1250) — compile-verified
//
#include <hip/hip_runtime.h>

// ---- LoRA fused forward for MI455X (gfx1250, wave32) -----------------------
// out[m, o] = SCALING * sum_r ( sum_k x[m,k] * A[r,k] ) * B[o,r]
// Memory-bound: 268 MB of HBM traffic @ 23.3 TB/s ~= 11.5 us floor.
// Fused single kernel: x read once, out written once, h kept in LDS.
// Rank-4 output projection maps exactly onto V_WMMA_F32_16X16X4_F32.

typedef float v2f __attribute__((ext_vector_type(2)));
typedef float v8f __attribute__((ext_vector_type(8)));

#define IN_F   1024
#define OUT_F  1024
#define RANK   4
#define SCALING 0.25f

#define ROWS_PER_BLOCK 64
#define THREADS        256
#define WAVES          (THREADS / 32)
#define ROWS_PER_WAVE  (ROWS_PER_BLOCK / WAVES)   // 8

__global__ __launch_bounds__(THREADS)
void lora_fused_kernel(const float* __restrict__ x,     // [M, 1024]
                       const float* __restrict__ A,     // [4, 1024]
                       const float* __restrict__ B,     // [1024, 4]
                       float* __restrict__ out,         // [M, 1024]
                       int m_total)
{
    // LDS: A transposed+prescaled (16KB), B copy (16KB), h tile (1KB)
    __shared__ float At[IN_F * RANK];          // At[k*4 + r] = A[r][k] * SCALING
    __shared__ float Bs[OUT_F * RANK];         // Bs[o*4 + r] = B[o][r]
    __shared__ float Hs[ROWS_PER_BLOCK * RANK];

    const int tid  = threadIdx.x;
    const int lane = tid & 31;
    const int wave = tid >> 5;
    const int row0 = blockIdx.x * ROWS_PER_BLOCK;

    // ---- Phase 0: stage A (transposed, scaled) and B into LDS --------------
    {
        // thread t covers k = 4t..4t+3 for each of the 4 A rows
        const int k0 = tid * 4;
        #pragma unroll
        for (int r = 0; r < RANK; ++r) {
            float4 v = *(const float4*)(A + (size_t)r * IN_F + k0);
            At[(k0 + 0) * RANK + r] = v.x * SCALING;
            At[(k0 + 1) * RANK + r] = v.y * SCALING;
            At[(k0 + 2) * RANK + r] = v.z * SCALING;
            At[(k0 + 3) * RANK + r] = v.w * SCALING;
        }
        // straight copy of B: 4096 floats = 1024 float4, 4 per thread
        #pragma unroll
        for (int i = 0; i < 4; ++i) {
            const int idx = i * THREADS + tid;
            ((float4*)Bs)[idx] = ((const float4*)B)[idx];
        }
    }
    __syncthreads();

    // ---- Phase 1: Hs = (x_tile @ A^T) * SCALING ----------------------------
    // One wave per row: lane-coalesced b32 loads of x, ds_load_b128 of At[k],
    // 4 FMAs per element, then 5-step xor-shuffle reduction across 32 lanes.
    for (int rr = 0; rr < ROWS_PER_WAVE; ++rr) {
        const int rowl = wave * ROWS_PER_WAVE + rr;
        const int rowg = row0 + rowl;
        if (rowg < m_total) {
            const float* __restrict__ xr = x + (size_t)rowg * IN_F;
            float a0 = 0.f, a1 = 0.f, a2 = 0.f, a3 = 0.f;
            #pragma unroll 4
            for (int i = 0; i < IN_F / 32; ++i) {
                const int k = i * 32 + lane;
                const float  xv = xr[k];
                const float4 av = *(const float4*)(At + k * RANK);
                a0 += xv * av.x;
                a1 += xv * av.y;
                a2 += xv * av.z;
                a3 += xv * av.w;
            }
            #pragma unroll
            for (int off = 16; off > 0; off >>= 1) {
                a0 += __shfl_xor(a0, off, 32);
                a1 += __shfl_xor(a1, off, 32);
                a2 += __shfl_xor(a2, off, 32);
                a3 += __shfl_xor(a3, off, 32);
            }
            if (lane == 0) {
                *(float4*)(Hs + rowl * RANK) = make_float4(a0, a1, a2, a3);
            }
        }
    }
    __syncthreads();

    // ---- Phase 2: out_tile = Hs @ B^T via V_WMMA_F32_16X16X4_F32 -----------
    // Wave w owns row sub-tile ms = (w/2)*16 and half of the 1024 output cols.
    // A operand (h fragment) is loaded once per wave, reused for 32 WMMAs.
    const int ms   = (wave >> 1) * 16;          // 0,16,32,48
    const int ch   = (wave & 1) * (OUT_F / 2);  // 0 or 512
    const int lrow = lane & 15;
    const int hi   = lane >> 4;                 // 0: K=0,1 / M=v ; 1: K=2,3 / M=v+8
    const int r0   = hi * 2;

    // A-matrix 16x4 f32 layout: lanes 0-15 -> {K=0,K=1}, lanes 16-31 -> {K=2,K=3}
    v2f afrag;
    {
        const float* hp = Hs + (ms + lrow) * RANK + r0;
        afrag[0] = hp[0];
        afrag[1] = hp[1];
    }

    for (int ci = 0; ci < (OUT_F / 2) / 16; ++ci) {
        const int n0 = ch + ci * 16;
        // B-matrix 4x16 f32 layout mirrors A: row K striped across VGPRs,
        // lanes 0-15 -> K=0,1 at column n0+lane, lanes 16-31 -> K=2,3.
        v2f bfrag;
        {
            const float* bp = Bs + (n0 + lrow) * RANK + r0;
            bfrag[0] = bp[0];
            bfrag[1] = bp[1];
        }
        v8f cz = {};
        // 8 args: (neg_a, A, neg_b, B, c_mod, C, reuse_a, reuse_b)
        v8f d = __builtin_amdgcn_wmma_f32_16x16x4_f32(
            false, afrag, false, bfrag, (short)0, cz, false, false);

        // C/D layout: VGPR v -> M = v (lanes 0-15) / M = v+8 (lanes 16-31), N = lane%16
        float* op = out + (size_t)(row0 + ms + hi * 8) * OUT_F + n0 + lrow;
        #pragma unroll
        for (int v = 0; v < 8; ++v) {
            *op = d[v];
            op += OUT_F;
        }
    }
}

extern "C" void kernel_launch(void* const* d_in, const int* in_sizes, int n_in,
                              void* d_out, int out_size, void* d_ws, size_t ws_size,
                              hipStream_t stream)
{
    const float* x = (const float*)d_in[0];   // [4*8192, 1024] flattened
    const float* A = (const float*)d_in[1];   // [4, 1024]
    const float* B = (const float*)d_in[2];   // [1024, 4]
    float* out = (float*)d_out;

    const int m_total = in_sizes[0] / IN_F;   // 32768
    const int grid = (m_total + ROWS_PER_BLOCK - 1) / ROWS_PER_BLOCK;

    lora_fused_kernel<<<grid, THREADS, 0, stream>>>(x, A, B, out, m_total);
}